// HmmNet_77936476553903
// MI455X (gfx1250) — compile-verified
//
#include <hip/hip_runtime.h>
#include <cstdint>

// HMM forward scan for MI455X (gfx1250, wave32).
// One wave per batch element; lane = hidden state (b == 32 == wave width).
// Per-timestep operands stream into LDS via gfx1250 async-to-LDS ops
// (ASYNCcnt), double-buffered in 16-step stages.
// Recurrence runs in log2 domain (native v_exp_f32/v_log_f32) with a
// carried-lse shift replacing the per-step max reduction (shift cancels
// exactly; fp32 exp2 range gives ~2x headroom over worst-case drift).

namespace {
constexpr int kBatch = 64;     // B
constexpr int kT     = 1024;   // T
constexpr int kT1    = 1025;   // T + 1
constexpr int kH     = 32;     // hidden options b (== lanes of wave32)
constexpr int kA     = 64;     // actions A
constexpr int kTB    = 16;     // timesteps per LDS stage
constexpr int kNBlk  = 64;     // stages covering t = 1 .. 1024 (t=1024 junk-but-safe)

constexpr float kLog2e = 1.44269504088896340736f;  // log2(e)
constexpr float kLn2   = 0.69314718055994530942f;  // ln(2)
}

__device__ __forceinline__ void async_load_b32(uint32_t lds_off, const void* gptr) {
  asm volatile("global_load_async_to_lds_b32 %0, %1, off"
               :: "v"(lds_off), "v"(gptr) : "memory");
}
__device__ __forceinline__ void async_load_b64(uint32_t lds_off, const void* gptr) {
  asm volatile("global_load_async_to_lds_b64 %0, %1, off"
               :: "v"(lds_off), "v"(gptr) : "memory");
}

__device__ __forceinline__ float wave_max32(float v) {
  #pragma unroll
  for (int off = 16; off > 0; off >>= 1) v = fmaxf(v, __shfl_xor(v, off, 32));
  return v;
}
__device__ __forceinline__ float wave_sum32(float v) {
  #pragma unroll
  for (int off = 16; off > 0; off >>= 1) v += __shfl_xor(v, off, 32);
  return v;
}
// Exact logsumexp (base-2) over the 32 lanes; used only in prologue/epilogue.
__device__ __forceinline__ float wave_lse2_exact(float v2) {
  const float m = wave_max32(v2);
  const float s = wave_sum32(exp2f(v2 - m));
  return m + log2f(s);
}

__global__ __launch_bounds__(32) void hmm_scan_kernel(
    const float*  __restrict__ action_logps,  // (B, T+1, 32, 64)
    const float2* __restrict__ stop_pairs,    // (B, T+1, 32, {STOP,CONT}) as float2
    const float*  __restrict__ start_logps,   // (B, T+1, 32)
    const int*    __restrict__ actions,       // (B, T)
    const int*    __restrict__ lengths,       // (B,)
    float*        __restrict__ partials)      // (B,)
{
  __shared__ float2 sbuf[2][kTB][kH];  // stop pair stream
  __shared__ float  tbuf[2][kTB][kH];  // start stream
  __shared__ float  abuf[2][kTB][kH];  // gathered action-logp stream

  const int Bb = blockIdx.x;
  const int h  = threadIdx.x;

  // Low 32 bits of a generic pointer to LDS == wave-relative LDS byte offset.
  const uint32_t sb = (uint32_t)(uintptr_t)&sbuf[0][0][0];
  const uint32_t tb = (uint32_t)(uintptr_t)&tbuf[0][0][0];
  const uint32_t ab = (uint32_t)(uintptr_t)&abuf[0][0][0];

  auto refill = [&](int buf, int tstart) {
    #pragma unroll
    for (int i = 0; i < kTB; ++i) {
      const int t  = tstart + i;
      const int tc = (t < kT) ? t : (kT - 1);              // clamp for actions[]
      const int a  = actions[(size_t)Bb * kT + tc];        // uniform -> SMEM load
      const size_t   row  = ((size_t)Bb * kT1 + (size_t)t) * kH + (size_t)h;
      const uint32_t slot = (uint32_t)((buf * kTB + i) * kH + h);
      async_load_b64(sb + slot * 8u, (const void*)(stop_pairs + row));
      async_load_b32(tb + slot * 4u, (const void*)(start_logps + row));
      const size_t arow =
          (((size_t)Bb * kT1 + (size_t)tc) * kH + (size_t)h) * kA + (size_t)a;
      async_load_b32(ab + slot * 4u, (const void*)(action_logps + arow));
    }
  };

  // t = 0: f0 = start_logps[:,0] + action_logps[:,0,:,a0]   (kept in log2 domain)
  const int a0 = actions[(size_t)Bb * kT];
  float f2 = (start_logps[(size_t)Bb * kT1 * kH + (size_t)h] +
              action_logps[(((size_t)Bb * kT1) * kH + (size_t)h) * kA + (size_t)a0]) *
             kLog2e;

  const int lenm1 = lengths[Bb] - 1;   // in [0, T-1]
  float x02 = f2;                      // correct snapshot if lenm1 == 0

  refill(0, 1);                        // stage 0 covers t = 1..16

  // Seed the uniform shift for the shifted logsumexp (exact once).
  float shift = wave_max32(f2);

  for (int k = 0; k < kNBlk; ++k) {
    const int cur = k & 1;
    if (k + 1 < kNBlk) {
      refill(cur ^ 1, 1 + kTB * (k + 1));       // prefetch next stage (48 asyncs)
      // async loads retire in order: <=48 outstanding => current stage landed
      asm volatile("s_wait_asynccnt 48" ::: "memory");
    } else {
      asm volatile("s_wait_asynccnt 0" ::: "memory");
    }
    const int tstart = 1 + kTB * k;
    #pragma unroll
    for (int i = 0; i < kTB; ++i) {
      const int t = tstart + i;                 // t = 1024 executes harmless junk
      const float2 sp   = sbuf[cur][i][h];
      const float beta2 = sp.x * kLog2e;        // STOP_IX   (off critical path)
      const float omb2  = sp.y * kLog2e;        // CONTINUE_IX
      const float st2   = tbuf[cur][i][h] * kLog2e;
      const float al2   = abuf[cur][i][h] * kLog2e;

      // Shifted logsumexp over hidden states; shift cancels exactly.
      // (beta2 - shift) is ready before f2 -> only add+exp2 on the chain.
      const float e   = exp2f(f2 + (beta2 - shift));
      const float s   = wave_sum32(e);
      const float lse = shift + log2f(s);

      const float r  = lse + st2;               // restart path
      const float u  = f2 + omb2;               // continue path (parallel w/ lse)
      const float mx = fmaxf(u, r);
      const float mn = fminf(u, r);
      f2 = mx + log2f(1.0f + exp2f(mn - mx)) + al2;

      x02   = (t == lenm1) ? f2 : x02;
      shift = lse;
    }
  }

  // total_logp = logsumexp_h( f[len-1] + stop[len, :, STOP] )  (exact, then -> ln)
  const int len = lenm1 + 1;
  const float x1 = stop_pairs[((size_t)Bb * kT1 + (size_t)len) * kH + (size_t)h].x;
  const float total = kLn2 * wave_lse2_exact(x02 + x1 * kLog2e);
  if (h == 0) partials[Bb] = total;
}

__global__ void hmm_finish_kernel(const float* __restrict__ partials,
                                  float* __restrict__ out) {
  if (blockIdx.x == 0 && threadIdx.x == 0) {
    float s = 0.0f;
    #pragma unroll
    for (int i = 0; i < kBatch; ++i) s += partials[i];   // fixed order: deterministic
    out[0] = -s;
  }
}

extern "C" void kernel_launch(void* const* d_in, const int* in_sizes, int n_in,
                              void* d_out, int out_size, void* d_ws, size_t ws_size,
                              hipStream_t stream) {
  const float*  action_logps = (const float*) d_in[0];
  const float2* stop_pairs   = (const float2*)d_in[1];
  const float*  start_logps  = (const float*) d_in[2];
  const int*    actions      = (const int*)   d_in[3];
  const int*    lengths      = (const int*)   d_in[4];
  float* partials = (float*)d_ws;

  hmm_scan_kernel<<<kBatch, kH, 0, stream>>>(action_logps, stop_pairs, start_logps,
                                             actions, lengths, partials);
  hmm_finish_kernel<<<1, 32, 0, stream>>>(partials, (float*)d_out);

  (void)in_sizes; (void)n_in; (void)out_size; (void)ws_size;
}